// DQNConv_50989851738775
// MI455X (gfx1250) — compile-verified
//
#include <hip/hip_runtime.h>
#include <cmath>

typedef __attribute__((ext_vector_type(2))) float v2f;
typedef __attribute__((ext_vector_type(4))) float v4f;
typedef __attribute__((ext_vector_type(8))) float v8f;

#define B_TOT 16384
#define OUTN  4096
#define KMOV  64
#define SPB   16     // samples per block (= WMMA M)
#define NTHR  256

// ---- repack W2 (64,32,3,3) -> B2[k=288][n=64], W3 (64,64,3,3) -> B3[k=576][n=64]
__global__ void repack_kernel(const float* __restrict__ W2,
                              const float* __restrict__ W3,
                              float* __restrict__ B2,
                              float* __restrict__ B3) {
  int i = blockIdx.x * blockDim.x + threadIdx.x;
  if (i < 288 * 64) {
    int k = i >> 6, n = i & 63;
    B2[i] = W2[n * 288 + k];
  } else if (i < 288 * 64 + 576 * 64) {
    int j = i - 288 * 64;
    int k = j >> 6, n = j & 63;
    B3[j] = W3[n * 576 + k];
  }
}

// ---- block reductions (wave32 shfl + 8-wave LDS combine)
__device__ __forceinline__ float blockRedMax(float v, float* red) {
#pragma unroll
  for (int o = 16; o > 0; o >>= 1) v = fmaxf(v, __shfl_xor(v, o, 32));
  if ((threadIdx.x & 31) == 0) red[threadIdx.x >> 5] = v;
  __syncthreads();
  if (threadIdx.x == 0) {
    float m = red[0];
#pragma unroll
    for (int i = 1; i < 8; ++i) m = fmaxf(m, red[i]);
    red[0] = m;
  }
  __syncthreads();
  float r = red[0];
  __syncthreads();
  return r;
}

__device__ __forceinline__ float blockRedSumF(float v, float* red) {
#pragma unroll
  for (int o = 16; o > 0; o >>= 1) v += __shfl_xor(v, o, 32);
  if ((threadIdx.x & 31) == 0) red[threadIdx.x >> 5] = v;
  __syncthreads();
  if (threadIdx.x == 0) {
    float m = 0.f;
#pragma unroll
    for (int i = 0; i < 8; ++i) m += red[i];
    red[0] = m;
  }
  __syncthreads();
  float r = red[0];
  __syncthreads();
  return r;
}

__device__ __forceinline__ int blockRedSumI(int v, int* red) {
#pragma unroll
  for (int o = 16; o > 0; o >>= 1) v += __shfl_xor(v, o, 32);
  if ((threadIdx.x & 31) == 0) red[threadIdx.x >> 5] = v;
  __syncthreads();
  if (threadIdx.x == 0) {
    int m = 0;
#pragma unroll
    for (int i = 0; i < 8; ++i) m += red[i];
    red[0] = m;
  }
  __syncthreads();
  int r = red[0];
  __syncthreads();
  return r;
}

__device__ __forceinline__ v8f wmma_f32(v2f a, v2f b, v8f c) {
  // D = A(16x4,f32) x B(4x16,f32) + C ; args: (neg_a, A, neg_b, B, c_mod, C, reuse_a, reuse_b)
  return __builtin_amdgcn_wmma_f32_16x16x4_f32(false, a, false, b, (short)0, c, false, false);
}

__global__ __launch_bounds__(NTHR)
void dqn_kernel(const float* __restrict__ x,
                const int*   __restrict__ pm,
                const float* __restrict__ W1,
                const float* __restrict__ b1,
                const float* __restrict__ b2,
                const float* __restrict__ b3,
                const float* __restrict__ B2w,
                const float* __restrict__ B3w,
                const float* __restrict__ Wl,
                const float* __restrict__ bl,
                float*       __restrict__ out) {
  __shared__ float sx[SPB * 49];                 // input tiles
  __shared__ float W1s[288];
  __shared__ float b1s[32];
  __shared__ float b2s[64];
  __shared__ float b3s[64];
  __shared__ int   offAs[288];                   // im2col offsets for conv2 A-frags
  __shared__ float A1s[SPB * 800];               // conv1 out: [s][c1][5*5]
  __shared__ __align__(16) float A2s[SPB * 576]; // conv2 out: [s][c2*9+p]
  __shared__ __align__(16) float feats[SPB * 64];
  __shared__ __align__(16) float scr[OUTN];      // per-sample softmax row
  __shared__ float redF[8];
  __shared__ int   redI[8];

  const int tid = threadIdx.x;
  const int s0  = blockIdx.x * SPB;

  // ---- phase 0: stage inputs / weights / tables
  for (int i = tid; i < SPB * 49; i += NTHR) sx[i] = x[s0 * 49 + i];
  for (int i = tid; i < 288; i += NTHR) {
    W1s[i] = W1[i];
    int c = i / 9, rem = i % 9;
    offAs[i] = c * 25 + (rem / 3) * 5 + (rem % 3);
  }
  if (tid < 32) b1s[tid] = b1[tid];
  if (tid < 64) { b2s[tid] = b2[tid]; b3s[tid] = b3[tid]; }
  __syncthreads();

  // ---- phase 1: conv1 (1->32, 7x7->5x5) on VALU, relu
  for (int o = tid; o < SPB * 800; o += NTHR) {
    int s = o / 800, rem = o % 800;
    int c = rem / 25, pix = rem % 25;
    int py = pix / 5, px = pix % 5;
    float acc = b1s[c];
    const float* xs = &sx[s * 49 + py * 7 + px];
    const float* w  = &W1s[c * 9];
#pragma unroll
    for (int r = 0; r < 3; ++r)
#pragma unroll
      for (int q = 0; q < 3; ++q)
        acc = fmaf(w[r * 3 + q], xs[r * 7 + q], acc);
    A1s[o] = fmaxf(acc, 0.0f);
  }
  __syncthreads();

  // ---- phase 2: conv2 via WMMA f32 16x16x4 (waves 0-3; one N=16 tile each)
  if (tid < 128) {
    const int wave = tid >> 5, lane = tid & 31;
    const int half = lane >> 4, lm = lane & 15;
    const int nloc = wave * 16 + lm;             // output channel this lane covers
    const int pb[9] = {0, 1, 2, 5, 6, 7, 10, 11, 12}; // oy*5+ox per output pos
    v8f acc[9];
#pragma unroll
    for (int p = 0; p < 9; ++p) acc[p] = (v8f){0.f,0.f,0.f,0.f,0.f,0.f,0.f,0.f};
    const float* Ab = &A1s[lm * 800];            // A row = sample lm
    for (int k0 = 0; k0 < 288; k0 += 4) {
      int k = k0 + half * 2;                     // this lane's K rows: k, k+1
      v2f b;
      b.x = B2w[(k + 0) * 64 + nloc];
      b.y = B2w[(k + 1) * 64 + nloc];
      int off0 = offAs[k], off1 = offAs[k + 1];
#pragma unroll
      for (int p = 0; p < 9; ++p) {
        v2f a;
        a.x = Ab[off0 + pb[p]];
        a.y = Ab[off1 + pb[p]];
        acc[p] = wmma_f32(a, b, acc[p]);
      }
    }
    float bias = b2s[nloc];
#pragma unroll
    for (int p = 0; p < 9; ++p)
#pragma unroll
      for (int r = 0; r < 8; ++r) {
        int s = r + half * 8;                    // D layout: M = r + 8*half
        A2s[s * 576 + nloc * 9 + p] = fmaxf(acc[p][r] + bias, 0.0f);
      }
  }
  __syncthreads();

  // ---- phase 3: conv3 via WMMA f32 16x16x4 (K = 64*9 = 576), relu -> feat
  if (tid < 128) {
    const int wave = tid >> 5, lane = tid & 31;
    const int half = lane >> 4, lm = lane & 15;
    const int nloc = wave * 16 + lm;
    v8f accf = (v8f){0.f,0.f,0.f,0.f,0.f,0.f,0.f,0.f};
    for (int k0 = 0; k0 < 576; k0 += 4) {
      int k = k0 + half * 2;
      v2f a = *(const v2f*)&A2s[lm * 576 + k];   // contiguous K in LDS
      v2f b;
      b.x = B3w[(k + 0) * 64 + nloc];
      b.y = B3w[(k + 1) * 64 + nloc];
      accf = wmma_f32(a, b, accf);
    }
    float bias = b3s[nloc];
#pragma unroll
    for (int r = 0; r < 8; ++r)
      feats[(r + half * 8) * 64 + nloc] = fmaxf(accf[r] + bias, 0.0f);
  }
  __syncthreads();

  // ---- phase 4: per-sample gathered logits + exact masked softmax + stream out
  for (int s = 0; s < SPB; ++s) {
    const int gs = s0 + s;
    for (int i = tid; i < OUTN; i += NTHR) scr[i] = -1000.0f;

    float vj = 0.0f; int idxj = 0;
    if (tid < KMOV) {
      idxj = pm[gs * KMOV + tid];
      float d = bl[idxj];
      const v4f* wr = (const v4f*)(Wl + (size_t)idxj * 64);
      const v4f* fr = (const v4f*)(feats + s * 64);
#pragma unroll
      for (int c = 0; c < 16; ++c) {
        v4f w = wr[c], f = fr[c];
        d = fmaf(w[0], f[0], d); d = fmaf(w[1], f[1], d);
        d = fmaf(w[2], f[2], d); d = fmaf(w[3], f[3], d);
      }
      vj = (d == 0.0f) ? -1000.0f : d;           // mask semantics: logit==0 -> -1000
    }
    __syncthreads();
    if (tid < KMOV) scr[idxj] = vj;              // duplicate idx write same value: benign
    __syncthreads();

    // pass 1: row max + count of non-background entries
    float lmx = -1000.0f; int lcnt = 0;
    {
      const int base = tid * 16;
#pragma unroll 4
      for (int i = 0; i < 16; ++i) {
        float v = scr[base + i];
        if (v != -1000.0f) { lmx = fmaxf(lmx, v); lcnt++; }
      }
    }
    float m  = blockRedMax(lmx, redF);
    int   cnt = blockRedSumI(lcnt, redI);
    float bg = expf(-1000.0f - m);               // background exp (0.0f unless m ~ -1000)

    // pass 2: sum of exp over non-background; cache exp back into scr
    float lsum = 0.0f;
    {
      const int base = tid * 16;
      for (int i = 0; i < 16; ++i) {
        float v = scr[base + i];
        if (v != -1000.0f) { float e = expf(v - m); scr[base + i] = e; lsum += e; }
      }
    }
    float se    = blockRedSumF(lsum, redF);
    float denom = se + (float)(OUTN - cnt) * bg;
    float inv   = 1.0f / denom;
    float bgOut = bg * inv;
    __syncthreads();

    // pass 3: stream the 16KB row out with non-temporal stores
    float* orow = out + (size_t)gs * OUTN;
    for (int i = tid; i < OUTN / 4; i += NTHR) {
      v4f v = *(const v4f*)&scr[i * 4];
      v4f r;
      r[0] = (v[0] == -1000.0f) ? bgOut : v[0] * inv;
      r[1] = (v[1] == -1000.0f) ? bgOut : v[1] * inv;
      r[2] = (v[2] == -1000.0f) ? bgOut : v[2] * inv;
      r[3] = (v[3] == -1000.0f) ? bgOut : v[3] * inv;
      __builtin_nontemporal_store(r, (v4f*)(orow + i * 4));
    }
    __syncthreads();
  }
}

extern "C" void kernel_launch(void* const* d_in, const int* in_sizes, int n_in,
                              void* d_out, int out_size, void* d_ws, size_t ws_size,
                              hipStream_t stream) {
  const float* x  = (const float*)d_in[0];
  const int*   pm = (const int*)  d_in[1];
  const float* W1 = (const float*)d_in[2];
  const float* b1 = (const float*)d_in[3];
  const float* W2 = (const float*)d_in[4];
  const float* b2 = (const float*)d_in[5];
  const float* W3 = (const float*)d_in[6];
  const float* b3 = (const float*)d_in[7];
  const float* Wl = (const float*)d_in[8];
  const float* bl = (const float*)d_in[9];
  float* out = (float*)d_out;

  float* B2 = (float*)d_ws;          // 288*64 floats
  float* B3 = B2 + 288 * 64;         // 576*64 floats  (total 221 KB of d_ws)

  const int repackN = 288 * 64 + 576 * 64;
  repack_kernel<<<(repackN + 255) / 256, 256, 0, stream>>>(W2, W3, B2, B3);
  dqn_kernel<<<B_TOT / SPB, NTHR, 0, stream>>>(x, pm, W1, b1, b2, b3, B2, B3, Wl, bl, out);
}